// ReadPath_10453950398508
// MI455X (gfx1250) — compile-verified
//
#include <hip/hip_runtime.h>
#include <hip/hip_bf16.h>

// ---------------------------------------------------------------------------
// Types for CDNA5 WMMA (gfx1250, wave32)
// ---------------------------------------------------------------------------
typedef __attribute__((ext_vector_type(2))) float v2f;
typedef __attribute__((ext_vector_type(4))) float v4f;
typedef __attribute__((ext_vector_type(8))) float v8f;
typedef __attribute__((ext_vector_type(4))) int v4i;

// Problem constants (from reference setup_inputs)
#define B_ 4
#define T_ 2048
#define H_ 2048
#define N_ 4096
#define D_ 256
#define NH_ 8
#define G_ 16
#define K_TOP 64
#define C_ (NH_ * D_)     // 2048
#define BT_ (B_ * T_)     // 8192
#define HG_ (H_ / 4)      // 512

__device__ __forceinline__ float wave_reduce_add(float v) {
#pragma unroll
    for (int off = 16; off; off >>= 1) v += __shfl_xor(v, off, 32);
    return v;
}

// ---------------------------------------------------------------------------
// CDNA5 async global->LDS copy (ASYNCcnt-tracked), 16B per lane.
// Builtin signature (from hipcc diagnostic): (v4i AS(1)*, v4i AS(3)*, Ii, Ii).
// ---------------------------------------------------------------------------
__device__ __forceinline__ void async_copy_b128(const float* __restrict__ g, float* l) {
#if __has_builtin(__builtin_amdgcn_global_load_async_to_lds_b128)
    __builtin_amdgcn_global_load_async_to_lds_b128(
        (__attribute__((address_space(1))) v4i*)(uintptr_t)g,
        (__attribute__((address_space(3))) v4i*)(unsigned)(uintptr_t)l, 0, 0);
#else
    unsigned lds = (unsigned)(uintptr_t)l;  // flat LDS addr low 32 bits = LDS byte offset
    asm volatile("global_load_async_to_lds_b128 %0, %1, off" ::"v"(lds), "v"(g)
                 : "memory");
#endif
}

__device__ __forceinline__ void wait_async0() {
#if __has_builtin(__builtin_amdgcn_s_wait_asynccnt)
    __builtin_amdgcn_s_wait_asynccnt(0);
#else
    asm volatile("s_wait_asynccnt 0x0" ::: "memory");
#endif
}

// ---------------------------------------------------------------------------
// Row L2-normalize: out[row, d] = in[row, d] / max(||in[row]||, 1e-8), D=256
// ---------------------------------------------------------------------------
__global__ __launch_bounds__(256) void rownorm_kernel(const float* __restrict__ in,
                                                      float* __restrict__ out) {
    int row = blockIdx.x;
    int tid = threadIdx.x;
    __shared__ float red[256];
    float v = in[(size_t)row * D_ + tid];
    red[tid] = v * v;
    __syncthreads();
    for (int st = 128; st; st >>= 1) {
        if (tid < st) red[tid] += red[tid + st];
        __syncthreads();
    }
    float r = fmaxf(sqrtf(red[0]), 1e-8f);
    out[(size_t)row * D_ + tid] = v / r;
}

// mean of goal angles over G=16 rows -> [256]
__global__ void meangoal_kernel(const float* __restrict__ ga, float* __restrict__ mg) {
    int d = threadIdx.x;
    float s = 0.f;
#pragma unroll
    for (int g = 0; g < G_; ++g) s += ga[g * D_ + d];
    mg[d] = s * (1.f / (float)G_);
}

// column mean of hidden over BT rows -> mean_hidden[2048]
__global__ __launch_bounds__(256) void colmean_kernel(const float* __restrict__ hidden,
                                                      float* __restrict__ mh) {
    int c = blockIdx.x * 256 + threadIdx.x;
    float s = 0.f;
    for (int r = 0; r < BT_; ++r) s += hidden[(size_t)r * H_ + c];
    mh[c] = s * (1.f / (float)BT_);
}

// out[row] = dot(mat[row, 0:cols], vec). One wave per row, 8 waves/block.
__global__ __launch_bounds__(256) void rowdot_kernel(const float* __restrict__ mat,
                                                     const float* __restrict__ vec,
                                                     float* __restrict__ out, int cols) {
    int row = blockIdx.x * 8 + (threadIdx.x >> 5);
    int lane = threadIdx.x & 31;
    const float* mr = mat + (size_t)row * cols;
    float p = 0.f;
    for (int i = lane; i < cols; i += 32) p += mr[i] * vec[i];
    p = wave_reduce_add(p);
    if (lane == 0) out[row] = p;
}

// rough_q[d] = mean over 8 heads of mean_q[h*256+d]
__global__ void roughq_kernel(const float* __restrict__ mq, float* __restrict__ rq) {
    int d = threadIdx.x;
    float s = 0.f;
#pragma unroll
    for (int h = 0; h < NH_; ++h) s += mq[h * D_ + d];
    rq[d] = s * (1.f / (float)NH_);
}

// Deterministic iterative top-64 argmax over 4096 scores (single block).
__global__ __launch_bounds__(256) void topk_kernel(const float* __restrict__ scores,
                                                   int* __restrict__ topk) {
    __shared__ float s[N_];
    __shared__ float rv[256];
    __shared__ int ri[256];
    int tid = threadIdx.x;
    for (int i = tid; i < N_; i += 256) s[i] = scores[i];
    __syncthreads();
    for (int it = 0; it < K_TOP; ++it) {
        float bv = -3.4e38f;
        int bi = 0x7fffffff;
        for (int i = tid; i < N_; i += 256) {
            float v = s[i];
            if (v > bv) { bv = v; bi = i; }   // ascending scan => lowest index on ties
        }
        rv[tid] = bv;
        ri[tid] = bi;
        __syncthreads();
        for (int st = 128; st; st >>= 1) {
            if (tid < st) {
                if (rv[tid + st] > rv[tid] ||
                    (rv[tid + st] == rv[tid] && ri[tid + st] < ri[tid])) {
                    rv[tid] = rv[tid + st];
                    ri[tid] = ri[tid + st];
                }
            }
            __syncthreads();
        }
        if (tid == 0) {
            topk[it] = ri[0];
            s[ri[0]] = -3.4e38f;
        }
        __syncthreads();
    }
}

// keys[k] = angles[topk[k]], values[k] = beliefs[topk[k]]
__global__ void gather_kernel(const int* __restrict__ topk, const float* __restrict__ angles,
                              const float* __restrict__ beliefs, float* __restrict__ keys,
                              float* __restrict__ values) {
    int k = blockIdx.x, d = threadIdx.x;
    int idx = topk[k];
    keys[k * D_ + d] = angles[(size_t)idx * D_ + d];
    values[k * D_ + d] = beliefs[(size_t)idx * D_ + d];
}

// max_goal_sim[k] = max_g (keys[k] . goal_angles[g]) * prio[g]; wave per k
__global__ __launch_bounds__(256) void goalsim_kernel(const float* __restrict__ keys,
                                                      const float* __restrict__ ga,
                                                      const float* __restrict__ prio,
                                                      float* __restrict__ mgs) {
    int k = blockIdx.x * 8 + (threadIdx.x >> 5);
    int lane = threadIdx.x & 31;
    float best = -3.4e38f;
    for (int g = 0; g < G_; ++g) {
        float p = 0.f;
#pragma unroll
        for (int i = 0; i < 8; ++i)
            p += keys[k * D_ + lane + 32 * i] * ga[g * D_ + lane + 32 * i];
        p = wave_reduce_add(p);
        best = fmaxf(best, p * prio[g]);
    }
    if (lane == 0) mgs[k] = best;
}

// temp[h], gate_w[h], goal_bias[h][k]
__global__ __launch_bounds__(512) void setup_kernel(const float* __restrict__ Wgoal,
                                                    const float* __restrict__ log_temp,
                                                    const float* __restrict__ mgoal,
                                                    const float* __restrict__ mgs,
                                                    float* __restrict__ temp,
                                                    float* __restrict__ goal_bias) {
    __shared__ float s_gw[NH_];
    int tid = threadIdx.x;
    if (tid < NH_) {
        temp[tid] = fmaxf(__expf(log_temp[tid] + 0.1f), 0.1f);
        float s = 0.f;
        for (int d = 0; d < D_; ++d) s += Wgoal[tid * D_ + d] * mgoal[d];
        s_gw[tid] = 1.f / (1.f + __expf(-s));
    }
    __syncthreads();
    int h = tid >> 6, k = tid & 63;
    goal_bias[tid] = s_gw[h] * mgs[k];
}

// ---------------------------------------------------------------------------
// WMMA fp32 GEMM: C[m,n] = sum_k A[m,k] * W[n,k]  (+bias[n]) (*rowScale[m])
// Block tile 128x64, BK=32, 8 waves, wave owns 16 rows x 64 cols (4x v8f acc).
// Double-buffered LDS fed by GLOBAL_LOAD_ASYNC_TO_LDS_B128 (ASYNCcnt),
// WMMA via V_WMMA_F32_16X16X4_F32. Requires M%128==0, N%64==0, K%32==0.
// ---------------------------------------------------------------------------
#define BM 128
#define BN 64
#define BK 32
#define LDS_S (BK + 4)  // padded row stride (floats); 36*4B keeps 16B alignment

__global__ __launch_bounds__(256) void gemm_wmma_f32(const float* __restrict__ A,
                                                     const float* __restrict__ W,
                                                     const float* __restrict__ bias,
                                                     const float* __restrict__ rowScale,
                                                     float* __restrict__ C, int M, int N,
                                                     int K) {
    __shared__ float As[2][BM * LDS_S];
    __shared__ float Ws[2][BN * LDS_S];

    const int tid = threadIdx.x;
    const int lane = tid & 31;
    const int wv = tid >> 5;
    const int r = lane & 15;
    const int hi = lane >> 4;

    const int nb = N / BN;
    const int m0 = (blockIdx.x / nb) * BM;
    const int n0 = (blockIdx.x % nb) * BN;

    v8f acc[4] = {};

    // Per-thread copy tasks (fixed across chunks): A = 4x b128, W = 2x b128.
    const int rowA[4] = {(tid + 0) >> 3, (tid + 256) >> 3, (tid + 512) >> 3,
                         (tid + 768) >> 3};
    const int rowW[2] = {(tid + 0) >> 3, (tid + 256) >> 3};
    const int vA = tid & 7;  // same sub-column for all tasks of this thread

    const int nchunks = K / BK;

    // Issue async copies for chunk `c` into buffer `buf`.
    auto issue = [&](int c, int buf) {
        int k0 = c * BK;
#pragma unroll
        for (int i = 0; i < 4; ++i)
            async_copy_b128(A + (size_t)(m0 + rowA[i]) * K + k0 + vA * 4,
                            &As[buf][rowA[i] * LDS_S + vA * 4]);
#pragma unroll
        for (int i = 0; i < 2; ++i)
            async_copy_b128(W + (size_t)(n0 + rowW[i]) * K + k0 + vA * 4,
                            &Ws[buf][rowW[i] * LDS_S + vA * 4]);
    };

    issue(0, 0);

    for (int c = 0; c < nchunks; ++c) {
        wait_async0();      // this wave's copies for chunk c have landed in LDS
        __syncthreads();    // all waves' copies landed; fences compute(c-1) reads too
        if (c + 1 < nchunks) issue(c + 1, (c + 1) & 1);  // overlap DMA with WMMA

        const int buf = c & 1;
        const float* arow = &As[buf][(wv * 16 + r) * LDS_S + 2 * hi];
        const float* wrow = &Ws[buf][r * LDS_S + 2 * hi];
#pragma unroll
        for (int kk = 0; kk < BK; kk += 4) {
            // A 16x4 fragment: lanes 0-15 K={kk,kk+1}, lanes 16-31 K={kk+2,kk+3}
            v2f a = *(const v2f*)(arow + kk);
#pragma unroll
            for (int nt = 0; nt < 4; ++nt) {
                // B 4x16 fragment: column n = r, mirrored K split via hi
                v2f b = *(const v2f*)(wrow + nt * 16 * LDS_S + kk);
                acc[nt] = __builtin_amdgcn_wmma_f32_16x16x4_f32(
                    false, a, false, b, (short)0, acc[nt], false, false);
            }
        }
    }

    // Epilogue: C/D layout — VGPR j holds M = j + 8*hi, N = n0 + nt*16 + r
    const int mbase = m0 + wv * 16 + 8 * hi;
#pragma unroll
    for (int nt = 0; nt < 4; ++nt) {
        int col = n0 + nt * 16 + r;
        float bv = bias ? bias[col] : 0.f;
#pragma unroll
        for (int j = 0; j < 8; ++j) {
            int m = mbase + j;
            float vv = acc[nt][j] + bv;
            if (rowScale) vv *= rowScale[m];
            C[(size_t)m * N + col] = vv;
        }
    }
}

// ---------------------------------------------------------------------------
// Read gate epilogue: gate[row] = sigmoid(signed_sqrt(relu(g1) . Wg2 + bg2))
// ---------------------------------------------------------------------------
__global__ __launch_bounds__(256) void gate_kernel(const float* __restrict__ g1,
                                                   const float* __restrict__ Wg2,
                                                   const float* __restrict__ bg2,
                                                   float* __restrict__ gate) {
    int row = blockIdx.x * 8 + (threadIdx.x >> 5);
    int lane = threadIdx.x & 31;
    float p = 0.f;
    for (int i = lane; i < HG_; i += 32) {
        float x = fmaxf(g1[(size_t)row * HG_ + i], 0.f);
        p += x * Wg2[i];
    }
    p = wave_reduce_add(p);
    if (lane == 0) {
        float g = p + bg2[0];
        float sg = (g > 0.f) ? 1.f : ((g < 0.f) ? -1.f : 0.f);
        g = sg * sqrtf(fmaxf(fabsf(g), 1e-6f));
        gate[row] = 1.f / (1.f + __expf(-g));
    }
}

// ---------------------------------------------------------------------------
// Attention: scores -> softmax -> attn_out + retrieved. Wave per (bt, head).
// ---------------------------------------------------------------------------
__global__ __launch_bounds__(256) void attn_kernel(const float* __restrict__ Q,
                                                   const float* __restrict__ keys,
                                                   const float* __restrict__ values,
                                                   const float* __restrict__ temp,
                                                   const float* __restrict__ goal_bias,
                                                   float* __restrict__ attn_out,
                                                   float* __restrict__ retr_out) {
    const int bt = blockIdx.x;
    const int head = threadIdx.x >> 5;
    const int lane = threadIdx.x & 31;
    __shared__ float s_attn[NH_][K_TOP];

    const float* q = Q + (size_t)bt * C_ + head * D_;
    float qr[8];
#pragma unroll
    for (int i = 0; i < 8; ++i) qr[i] = q[lane + 32 * i];

    const float tscale = temp[head] * 0.0625f;  // temp / sqrt(256)

    for (int k = 0; k < K_TOP; ++k) {
        const float* kr = keys + k * D_;
        float p = 0.f;
#pragma unroll
        for (int i = 0; i < 8; ++i) p += qr[i] * kr[lane + 32 * i];
        p = wave_reduce_add(p);
        if (lane == 0) s_attn[head][k] = p * tscale + goal_bias[head * K_TOP + k];
    }
    __syncthreads();

    float s0 = s_attn[head][lane];
    float s1 = s_attn[head][lane + 32];
    float mx = fmaxf(s0, s1);
#pragma unroll
    for (int off = 16; off; off >>= 1) mx = fmaxf(mx, __shfl_xor(mx, off, 32));
    float e0 = __expf(s0 - mx), e1 = __expf(s1 - mx);
    float sum = wave_reduce_add(e0 + e1);
    float inv = 1.f / sum;
    e0 *= inv;
    e1 *= inv;
    s_attn[head][lane] = e0;
    s_attn[head][lane + 32] = e1;
    size_t abase = ((size_t)bt * NH_ + head) * K_TOP;
    attn_out[abase + lane] = e0;
    attn_out[abase + lane + 32] = e1;
    __syncthreads();

#pragma unroll
    for (int i = 0; i < 8; ++i) {
        int d = lane + 32 * i;
        float acc = 0.f;
        for (int k = 0; k < K_TOP; ++k) acc += s_attn[head][k] * values[k * D_ + d];
        retr_out[((size_t)bt * NH_ + head) * D_ + d] = acc;
    }
}

// ---------------------------------------------------------------------------
// flat = retrieved + silu(depthwise_causal_conv4(retrieved))
// ---------------------------------------------------------------------------
__global__ __launch_bounds__(256) void conv_kernel(const float* __restrict__ retr,
                                                   const float* __restrict__ conv_w,
                                                   float* __restrict__ flat) {
    size_t idx = (size_t)blockIdx.x * 256 + threadIdx.x;  // < BT_*C_
    int c = (int)(idx & (C_ - 1));
    size_t row = idx >> 11;  // bt index
    int t = (int)(row & (T_ - 1));
    const float* w = conv_w + c * 4;
    float acc = 0.f;
#pragma unroll
    for (int j = 0; j < 4; ++j) {
        int tt = t - 3 + j;
        if (tt >= 0) acc += retr[idx + ((size_t)(j - 3)) * C_] * w[j];
    }
    float s = acc / (1.f + __expf(-acc));  // SiLU
    flat[idx] = retr[idx] + s;
}

// ---------------------------------------------------------------------------
// Host launcher
// ---------------------------------------------------------------------------
extern "C" void kernel_launch(void* const* d_in, const int* in_sizes, int n_in,
                              void* d_out, int out_size, void* d_ws, size_t ws_size,
                              hipStream_t stream) {
    (void)in_sizes; (void)n_in; (void)out_size; (void)ws_size;

    const float* hidden   = (const float*)d_in[0];   // [4,2048,2048]
    const float* beliefs  = (const float*)d_in[1];   // [4096,256]
    const float* goals    = (const float*)d_in[2];   // [16,256]
    const float* prio     = (const float*)d_in[3];   // [16]
    const float* Wq       = (const float*)d_in[4];   // [2048,2048]
    const float* Wo       = (const float*)d_in[5];   // [2048,2048]
    const float* Wgoal    = (const float*)d_in[6];   // [8,256]
    const float* log_temp = (const float*)d_in[7];   // [8]
    const float* Wu       = (const float*)d_in[8];   // [2048,2048]
    const float* Wg1      = (const float*)d_in[9];   // [512,2048]
    const float* bg1      = (const float*)d_in[10];  // [512]
    const float* Wg2      = (const float*)d_in[11];  // [1,512]
    const float* bg2      = (const float*)d_in[12];  // [1]
    const float* conv_w   = (const float*)d_in[13];  // [2048,1,4]

    float* out  = (float*)d_out;                         // [8192,2048]
    float* util = out + (size_t)BT_ * H_;                // [8192,2048]
    float* attn = util + (size_t)BT_ * H_;               // [8192,8,64]
    float* retr = attn + (size_t)BT_ * NH_ * K_TOP;      // [8192,8,256]

    float* w = (float*)d_ws;
    float* angles = w;  w += (size_t)N_ * D_;
    float* gang   = w;  w += (size_t)G_ * D_;
    float* mgoal  = w;  w += D_;
    float* mhid   = w;  w += H_;
    float* meanq  = w;  w += C_;
    float* roughq = w;  w += D_;
    float* roughs = w;  w += N_;
    float* keys   = w;  w += (size_t)K_TOP * D_;
    float* values = w;  w += (size_t)K_TOP * D_;
    float* mgs    = w;  w += K_TOP;
    float* temp   = w;  w += NH_;
    float* gbias  = w;  w += NH_ * K_TOP;
    float* gate   = w;  w += BT_;
    float* Qb     = w;  w += (size_t)BT_ * C_;
    float* g1     = w;  w += (size_t)BT_ * HG_;
    float* flat   = w;  w += (size_t)BT_ * C_;
    int*   topk   = (int*)w;

    // --- selection / setup path ---
    rownorm_kernel<<<N_, 256, 0, stream>>>(beliefs, angles);
    rownorm_kernel<<<G_, 256, 0, stream>>>(goals, gang);
    meangoal_kernel<<<1, 256, 0, stream>>>(gang, mgoal);
    colmean_kernel<<<H_ / 256, 256, 0, stream>>>(hidden, mhid);
    rowdot_kernel<<<C_ / 8, 256, 0, stream>>>(Wq, mhid, meanq, H_);
    roughq_kernel<<<1, 256, 0, stream>>>(meanq, roughq);
    rowdot_kernel<<<N_ / 8, 256, 0, stream>>>(angles, roughq, roughs, D_);
    topk_kernel<<<1, 256, 0, stream>>>(roughs, topk);
    gather_kernel<<<K_TOP, 256, 0, stream>>>(topk, angles, beliefs, keys, values);
    goalsim_kernel<<<K_TOP / 8, 256, 0, stream>>>(keys, gang, prio, mgs);
    setup_kernel<<<1, 512, 0, stream>>>(Wgoal, log_temp, mgoal, mgs, temp, gbias);

    // --- heavy GEMMs (fp32 WMMA, async double-buffered LDS) ---
    gemm_wmma_f32<<<(BT_ / BM) * (C_ / BN), 256, 0, stream>>>(
        hidden, Wq, nullptr, nullptr, Qb, BT_, C_, H_);
    gemm_wmma_f32<<<(BT_ / BM) * (HG_ / BN), 256, 0, stream>>>(
        hidden, Wg1, bg1, nullptr, g1, BT_, HG_, H_);
    gate_kernel<<<BT_ / 8, 256, 0, stream>>>(g1, Wg2, bg2, gate);

    // --- attention core ---
    attn_kernel<<<BT_, 256, 0, stream>>>(Qb, keys, values, temp, gbias, attn, retr);

    // --- conv fusion + output GEMMs ---
    conv_kernel<<<(BT_ * C_) / 256, 256, 0, stream>>>(retr, conv_w, flat);
    gemm_wmma_f32<<<(BT_ / BM) * (H_ / BN), 256, 0, stream>>>(
        flat, Wo, nullptr, gate, out, BT_, H_, C_);
    gemm_wmma_f32<<<(BT_ / BM) * (H_ / BN), 256, 0, stream>>>(
        flat, Wu, nullptr, nullptr, util, BT_, H_, C_);
}